// MaxExtractor_52501680227023
// MI455X (gfx1250) — compile-verified
//
#include <hip/hip_runtime.h>
#include <hip/hip_bf16.h>
#include <math.h>

#define BATCH 8
#define NPRED 4096
#define MGT   2048
#define UNION 4096
#define PRED_PER_BLOCK 256
#define GT_TILE 512
#define THREADS 256

// ---- CDNA5 async global->LDS staging ---------------------------------------
// ROCm 7.2 clang-22 declares __builtin_amdgcn_global_load_async_to_lds_b128
// with strictly-typed pointers: (AS1 int4* src, AS3 int4* dst, imm off, imm cpol).
typedef int v4i __attribute__((vector_size(16)));
typedef __attribute__((address_space(1))) v4i gv4i;
typedef __attribute__((address_space(3))) v4i lv4i;

#if defined(__gfx1250__)
  #define ON_GFX1250 1
#else
  #define ON_GFX1250 0
#endif

#if ON_GFX1250 && __has_builtin(__builtin_amdgcn_global_load_async_to_lds_b128)
  // AS3 pointers are 32-bit LDS offsets == low 32 bits of the generic pointer
  // (LDS aperture lives in the high bits), so the integer round-trip is exact.
  #define ASYNC_COPY_B128(gptr, lptr)                                          \
    __builtin_amdgcn_global_load_async_to_lds_b128(                            \
        (gv4i*)(unsigned long long)(gptr),                                     \
        (lv4i*)(unsigned int)(unsigned long long)(const void*)(lptr),          \
        0, 0)
  #define HAVE_ASYNC_LDS 1
#elif ON_GFX1250
  // Fallback: raw CDNA5 mnemonic. vdst = 32-bit LDS byte address, vaddr = 64b
  // global address, SADDR=off (GV mode).
  #define ASYNC_COPY_B128(gptr, lptr)                                          \
    asm volatile("global_load_async_to_lds_b128 %0, %1, off"                   \
                 :: "v"((unsigned int)(unsigned long long)(const void*)(lptr)),\
                    "v"((unsigned long long)(gptr))                            \
                 : "memory")
  #define HAVE_ASYNC_LDS 1
#else
  #define HAVE_ASYNC_LDS 0
#endif

#if ON_GFX1250
  #if __has_builtin(__builtin_amdgcn_s_wait_asynccnt)
    #define ASYNC_WAIT() __builtin_amdgcn_s_wait_asynccnt(0)
  #else
    #define ASYNC_WAIT() asm volatile("s_wait_asynccnt 0" ::: "memory")
  #endif
#else
  #define ASYNC_WAIT()
#endif
// ---------------------------------------------------------------------------

__device__ __forceinline__ float wave32_max(float v) {
  // wave32 butterfly max
  #pragma unroll
  for (int off = 16; off > 0; off >>= 1)
    v = fmaxf(v, __shfl_xor(v, off, 32));
  return v;
}

// Main kernel first in the TU so the disasm snippet shows the async-LDS path.
__global__ __launch_bounds__(THREADS)
void iou_max_kernel(const float*  __restrict__ pred_boxes,
                    const int*    __restrict__ pred_classes,
                    const float*  __restrict__ gt_boxes,
                    int*          __restrict__ batch_max_bits) {
  __shared__ float4 s_gt[GT_TILE];          // 8 KB tile of GT boxes
  __shared__ float  s_wave[THREADS / 32];

  const int tid = threadIdx.x;
  const int b   = blockIdx.z;
  const int m0  = blockIdx.y * GT_TILE;
  const int n   = blockIdx.x * PRED_PER_BLOCK + tid;

  // ---- 1) kick off the GT-tile DMA into LDS (ASYNCcnt-tracked) ----
  const float4* gsrc = (const float4*)(gt_boxes + ((size_t)b * MGT + m0) * 4);
#if HAVE_ASYNC_LDS
  #pragma unroll
  for (int i = tid; i < GT_TILE; i += THREADS)
    ASYNC_COPY_B128(gsrc + i, &s_gt[i]);
#endif

  // ---- 2) per-thread pred fetches overlap the DMA ----
  const bool   is_person = (pred_classes[(size_t)b * NPRED + n] == 0);
  const float4 p         = ((const float4*)pred_boxes)[(size_t)b * NPRED + n];
  const float  area_p    = (p.z - p.x) * (p.w - p.y);

  // ---- 3) wait for DMA, publish tile to all waves ----
#if HAVE_ASYNC_LDS
  ASYNC_WAIT();
#else
  for (int i = tid; i < GT_TILE; i += THREADS) s_gt[i] = gsrc[i];
#endif
  __syncthreads();

  // ---- 4) masked IoU max over the tile ----
  // Track (inter, union) of the running best and compare by cross-
  // multiplication: one divide per THREAD instead of per PAIR.
  float best_inter = 0.0f;
  float best_uni   = 1.0f;

  if (is_person) {
    #pragma unroll 4
    for (int j = 0; j < GT_TILE; ++j) {
      const float4 g = s_gt[j];                       // ds_load_b128 broadcast
      float iw = fminf(p.z, g.z) - fmaxf(p.x, g.x);
      float ih = fminf(p.w, g.w) - fmaxf(p.y, g.y);
      iw = fmaxf(iw, 0.0f);
      ih = fmaxf(ih, 0.0f);
      const float inter  = iw * ih;                   // ==0 for zeroed GT rows
      const float area_g = (g.z - g.x) * (g.w - g.y);
      const float uni    = area_p + area_g - inter;   // > 0 always
      if (inter * best_uni > best_inter * uni) {      // inter/uni > best ratio
        best_inter = inter;
        best_uni   = uni;
      }
    }
  }
  const float my_iou = best_inter / fmaxf(best_uni, 1e-9f);

  // ---- 5) block reduction: wave32 shuffle, then cross-wave via LDS ----
  const float wmax = wave32_max(my_iou);
  if ((tid & 31) == 0) s_wave[tid >> 5] = wmax;
  __syncthreads();
  if (tid == 0) {
    float bm = s_wave[0];
    #pragma unroll
    for (int w = 1; w < THREADS / 32; ++w) bm = fmaxf(bm, s_wave[w]);
    // All candidates are >= 0 -> IEEE bits are monotone under signed-int max.
    atomicMax(&batch_max_bits[b], __float_as_int(bm));
  }
}

__global__ void iou_init_kernel(int* __restrict__ batch_max_bits) {
  if (threadIdx.x < BATCH) batch_max_bits[threadIdx.x] = 0;  // bits of +0.0f
}

__global__ void finalize_kernel(const float* __restrict__ union_scores,
                                const int*   __restrict__ union_classes,
                                const int*   __restrict__ batch_max_bits,
                                float*       __restrict__ out) {
  __shared__ float s_wave[4];
  const int tid = threadIdx.x;  // 128 threads
  float m = -INFINITY;
  for (int i = tid; i < UNION; i += 128) {
    const float sc = union_scores[i];
    m = (union_classes[i] == 0) ? fmaxf(m, sc) : m;
  }
  const float wmax = wave32_max(m);
  if ((tid & 31) == 0) s_wave[tid >> 5] = wmax;
  __syncthreads();
  if (tid == 0) {
    const float max_prob =
        fmaxf(fmaxf(s_wave[0], s_wave[1]), fmaxf(s_wave[2], s_wave[3]));
    float sum = 0.0f;
    #pragma unroll
    for (int b = 0; b < BATCH; ++b) sum += __int_as_float(batch_max_bits[b]);
    out[0] = max_prob;                 // max_prob_t
    out[1] = sum * (1.0f / BATCH);     // max_iou_t (mean over batches)
  }
}

extern "C" void kernel_launch(void* const* d_in, const int* in_sizes, int n_in,
                              void* d_out, int out_size, void* d_ws, size_t ws_size,
                              hipStream_t stream) {
  const float* pred_boxes    = (const float*)d_in[0];
  // d_in[1] = pred_scores: unused by the reference computation.
  const int*   pred_classes  = (const int*)d_in[2];
  const float* gt_boxes      = (const float*)d_in[3];
  const float* union_scores  = (const float*)d_in[4];
  const int*   union_classes = (const int*)d_in[5];
  float* out = (float*)d_out;
  int*   bmax = (int*)d_ws;  // 8 per-batch IoU-max accumulators (float bits)

  iou_init_kernel<<<1, 32, 0, stream>>>(bmax);

  dim3 grid(NPRED / PRED_PER_BLOCK, MGT / GT_TILE, BATCH);  // (16, 4, 8)
  iou_max_kernel<<<grid, THREADS, 0, stream>>>(pred_boxes, pred_classes,
                                               gt_boxes, bmax);

  finalize_kernel<<<1, 128, 0, stream>>>(union_scores, union_classes, bmax, out);
}